// Pointnet2Backbone_1511828488206
// MI455X (gfx1250) — compile-verified
//
#include <hip/hip_runtime.h>
#include <hip/hip_bf16.h>

typedef __attribute__((ext_vector_type(2))) float v2f;
typedef __attribute__((ext_vector_type(8))) float v8f;

#define BN_EPS 1e-5f
#define KCHUNK 64

// ---------------------------------------------------------------------------
// fill
// ---------------------------------------------------------------------------
__global__ void fill_kernel(float* p, long long n, float v) {
    long long i = (long long)blockIdx.x * blockDim.x + threadIdx.x;
    if (i < n) p[i] = v;
}

// ---------------------------------------------------------------------------
// Farthest point sampling: one block per batch, serial over npoint.
// ---------------------------------------------------------------------------
__global__ void fps_kernel(const float* __restrict__ xyz, int N, int stride,
                           int npoint, int* __restrict__ inds,
                           float* __restrict__ mind) {
    const int b = blockIdx.x;
    const int tid = threadIdx.x;
    const int T = blockDim.x;
    const float* P = xyz + (size_t)b * N * stride;
    float* md = mind + (size_t)b * N;
    int* out = inds + (size_t)b * npoint;

    __shared__ float sv[256];
    __shared__ int   si[256];
    __shared__ int   slast;

    for (int n = tid; n < N; n += T) md[n] = 1e10f;
    if (tid == 0) { out[0] = 0; slast = 0; }
    __syncthreads();

    for (int i = 1; i < npoint; ++i) {
        const int last = slast;
        const float lx = P[(size_t)last * stride + 0];
        const float ly = P[(size_t)last * stride + 1];
        const float lz = P[(size_t)last * stride + 2];
        float bv = -1.0f; int bi = 0;
        for (int n = tid; n < N; n += T) {
            const float dx = P[(size_t)n * stride + 0] - lx;
            const float dy = P[(size_t)n * stride + 1] - ly;
            const float dz = P[(size_t)n * stride + 2] - lz;
            const float d = dx * dx + dy * dy + dz * dz;
            float m = md[n];
            if (d < m) { m = d; md[n] = m; }
            if (m > bv) { bv = m; bi = n; }   // strict > keeps smallest index on ties
        }
        sv[tid] = bv; si[tid] = bi;
        __syncthreads();
        for (int off = 128; off > 0; off >>= 1) {
            if (tid < off) {
                const float ov = sv[tid + off]; const int oi = si[tid + off];
                if (ov > sv[tid] || (ov == sv[tid] && oi < si[tid])) {
                    sv[tid] = ov; si[tid] = oi;
                }
            }
            __syncthreads();
        }
        if (tid == 0) { out[i] = si[0]; slast = si[0]; }
        __syncthreads();
    }
}

// ---------------------------------------------------------------------------
// Gather sampled centroids -> new_xyz (B,S,3)
// ---------------------------------------------------------------------------
__global__ void gather_xyz_kernel(const float* __restrict__ xyz, int N, int stride,
                                  const int* __restrict__ inds, int S, int B,
                                  float* __restrict__ new_xyz) {
    int t = blockIdx.x * blockDim.x + threadIdx.x;
    if (t >= B * S) return;
    const int b = t / S;
    const int n = inds[t];
    const float* p = xyz + ((size_t)b * N + n) * stride;
    float* q = new_xyz + (size_t)t * 3;
    q[0] = p[0]; q[1] = p[1]; q[2] = p[2];
}

// ---------------------------------------------------------------------------
// Ball query: first nsample in-radius indices (increasing), padded with the
// first found; if none, all = N-1 (matches the reference's sort trick).
// ---------------------------------------------------------------------------
__global__ void ball_query_kernel(const float* __restrict__ xyz, int N, int stride,
                                  const float* __restrict__ new_xyz, int S, int B,
                                  float r2, int nsample, int* __restrict__ idx) {
    int t = blockIdx.x * blockDim.x + threadIdx.x;
    if (t >= B * S) return;
    const int b = t / S;
    const float* P = xyz + (size_t)b * N * stride;
    const float qx = new_xyz[(size_t)t * 3 + 0];
    const float qy = new_xyz[(size_t)t * 3 + 1];
    const float qz = new_xyz[(size_t)t * 3 + 2];
    int* o = idx + (size_t)t * nsample;
    int cnt = 0;
    for (int n = 0; n < N && cnt < nsample; ++n) {
        const float dx = P[(size_t)n * stride + 0] - qx;
        const float dy = P[(size_t)n * stride + 1] - qy;
        const float dz = P[(size_t)n * stride + 2] - qz;
        const float d2 = dx * dx + dy * dy + dz * dz;
        if (d2 < r2) {
            if (cnt == 0) for (int j = 0; j < nsample; ++j) o[j] = n;
            o[cnt++] = n;
        }
    }
    if (cnt == 0) for (int j = 0; j < nsample; ++j) o[j] = N - 1;
}

// ---------------------------------------------------------------------------
// Build X matrix (cpad rows x ncol cols, row-major): rows 0..2 = rel/radius,
// rows 3..3+Cf-1 = gathered feats, remaining rows zero.  col = (b*S+s)*K + k
// ---------------------------------------------------------------------------
__global__ void build_x_kernel(const float* __restrict__ xyz, int N, int stride,
                               const float* __restrict__ new_xyz,
                               const int* __restrict__ idx,
                               int S, int K, int B, float invR,
                               const float* __restrict__ fbase,
                               long long fBatch, long long fChan, long long fPt, int Cf,
                               float* __restrict__ X, int cpad) {
    const long long ncol = (long long)B * S * K;
    long long col = (long long)blockIdx.x * blockDim.x + threadIdx.x;
    if (col >= ncol) return;
    const long long t = col / K;            // b*S + s
    const int b = (int)(t / S);
    const int n = idx[col];
    const float* p = xyz + ((size_t)b * N + n) * stride;
    const float* q = new_xyz + (size_t)t * 3;
    X[col]              = (p[0] - q[0]) * invR;
    X[ncol + col]       = (p[1] - q[1]) * invR;
    X[2 * ncol + col]   = (p[2] - q[2]) * invR;
    for (int c = 0; c < Cf; ++c)
        X[(long long)(3 + c) * ncol + col] =
            fbase[(long long)b * fBatch + (long long)c * fChan + (long long)n * fPt];
    for (int c = 3 + Cf; c < cpad; ++c)
        X[(long long)c * ncol + col] = 0.0f;
}

// ---------------------------------------------------------------------------
// GEMM Y(O x ncol) = W(O x cin) * X(cpad x ncol) with fp32 WMMA 16x16x4.
// Register-blocked over M (compile-time MT tiles): each block computes MT
// 16-row tiles for its 64 columns, so each B fragment is loaded once and fed
// to MT WMMAs, and X is streamed from HBM exactly once per block.  Weight
// slab staged in LDS per 64-wide K chunk.  Per-channel sum / sum-of-squares
// accumulated for batch-stats BN.
// Block: 128 threads = 4 waves; grid.x = ncol/64, grid.y = (O/16)/MT.
// ---------------------------------------------------------------------------
template <int MT>
__global__ void gemm_bn_stats_kernel(const float* __restrict__ W, int cin, int cpad,
                                     const float* __restrict__ X, long long ncol,
                                     float* __restrict__ Y,
                                     float* __restrict__ csum,
                                     float* __restrict__ csq) {
    constexpr int MROWS = MT * 16;
    __shared__ float Wl[KCHUNK * 128];      // [kLocal][mLocal], stride 128 (32 KB)
    const int tid  = threadIdx.x;
    const int wave = tid >> 5;
    const int lane = tid & 31;
    const int half = lane >> 4;             // 0: lanes 0-15, 1: lanes 16-31
    const int mn   = lane & 15;
    const int mrow0 = blockIdx.y * MROWS;   // first output channel of this block
    const long long col0 = ((long long)blockIdx.x * 4 + wave) * 16;

    v8f acc[MT] = {};
    const float* xcol = X + col0 + mn;

    for (int k0 = 0; k0 < cpad; k0 += KCHUNK) {
        const int kc = (cpad - k0 < KCHUNK) ? (cpad - k0) : KCHUNK;
        // Stage MT*16 x kc weight slab into LDS (zero-pad channels >= cin).
        for (int i = tid; i < MROWS * kc; i += blockDim.x) {
            const int kl = i / MROWS, ml = i - kl * MROWS;
            const int c = k0 + kl;
            Wl[kl * 128 + ml] = (c < cin) ? W[(size_t)(mrow0 + ml) * cin + c] : 0.0f;
        }
        __syncthreads();

        for (int kl = 0; kl < kc; kl += 4) {
            const int kk = kl + 2 * half;
            v2f bf;
            const float* xp = xcol + (size_t)(k0 + kk) * ncol;
            bf.x = xp[0];                   // B: K=kk,   N=mn
            bf.y = xp[ncol];                // B: K=kk+1, N=mn
            if (kl + 4 < kc)
                __builtin_prefetch((const void*)(xp + 4 * (size_t)ncol), 0, 1);
#pragma unroll
            for (int t = 0; t < MT; ++t) {
                v2f a;
                a.x = Wl[kk * 128 + t * 16 + mn];        // A: M=mn, K=kk
                a.y = Wl[(kk + 1) * 128 + t * 16 + mn];  // A: M=mn, K=kk+1
                acc[t] = __builtin_amdgcn_wmma_f32_16x16x4_f32(
                    false, a, false, bf, (short)0, acc[t], false, false);
            }
        }
        __syncthreads();
    }

    // Store D (non-temporal: Y exceeds L2 on the big layers) + BN statistics.
#pragma unroll
    for (int t = 0; t < MT; ++t) {
#pragma unroll
        for (int r = 0; r < 8; ++r) {
            const int m = mrow0 + t * 16 + r + 8 * half;
            const float v = acc[t][r];
            __builtin_nontemporal_store(v, &Y[(size_t)m * ncol + col0 + mn]);
            float sv = v, qv = v * v;
#pragma unroll
            for (int off = 1; off < 16; off <<= 1) {
                sv += __shfl_xor(sv, off, 32);
                qv += __shfl_xor(qv, off, 32);
            }
            if (mn == 0) {
                atomicAdd(&csum[m], sv);
                atomicAdd(&csq[m], qv);
            }
        }
    }
}

// ---------------------------------------------------------------------------
// BN finalize: per-channel scale/shift from accumulated stats.
// ---------------------------------------------------------------------------
__global__ void bn_finalize_kernel(const float* __restrict__ csum,
                                   const float* __restrict__ csq,
                                   const float* __restrict__ gamma,
                                   const float* __restrict__ beta,
                                   int O, float inv_count,
                                   float* __restrict__ scale,
                                   float* __restrict__ shift) {
    int o = blockIdx.x * blockDim.x + threadIdx.x;
    if (o >= O) return;
    const float mu  = csum[o] * inv_count;
    const float var = csq[o] * inv_count - mu * mu;
    const float sc  = gamma[o] * rsqrtf(var + BN_EPS);
    scale[o] = sc;
    shift[o] = beta[o] - mu * sc;
}

// ---------------------------------------------------------------------------
// Apply BN + ReLU in place.  grid: (ncol/256, O)
// ---------------------------------------------------------------------------
__global__ void apply_bn_relu_kernel(float* __restrict__ Y, long long ncol,
                                     const float* __restrict__ scale,
                                     const float* __restrict__ shift) {
    const int o = blockIdx.y;
    const long long i = (size_t)o * ncol + (long long)blockIdx.x * blockDim.x + threadIdx.x;
    const float v = Y[i] * scale[o] + shift[o];
    Y[i] = v > 0.0f ? v : 0.0f;
}

// ---------------------------------------------------------------------------
// Max-pool over the K neighbors -> feats (B,C,S)
// ---------------------------------------------------------------------------
__global__ void maxpool_kernel(const float* __restrict__ Y, int B, int C, int S, int K,
                               float* __restrict__ feats) {
    int t = blockIdx.x * blockDim.x + threadIdx.x;
    if (t >= B * C * S) return;
    const int s = t % S;
    const int c = (t / S) % C;
    const int b = t / (S * C);
    const long long ncol = (long long)B * S * K;
    const float* p = Y + (size_t)c * ncol + ((size_t)b * S + s) * K;
    float m = p[0];
    for (int k = 1; k < K; ++k) m = fmaxf(m, p[k]);
    feats[t] = m;
}

// ---------------------------------------------------------------------------
// Final output: [new_xyz4 (B,S4,3)] ++ [mean_s feats4 (B,C4)]
// ---------------------------------------------------------------------------
__global__ void final_out_kernel(const float* __restrict__ nx,
                                 const float* __restrict__ f4,
                                 int B, int S, int C, float* __restrict__ out) {
    int t = blockIdx.x * blockDim.x + threadIdx.x;
    const int nxyz = B * S * 3;
    if (t < nxyz) { out[t] = nx[t]; return; }
    t -= nxyz;
    if (t >= B * C) return;
    const int b = t / C, c = t % C;
    const float* p = f4 + ((size_t)b * C + c) * S;
    float s = 0.0f;
    for (int i = 0; i < S; ++i) s += p[i];
    out[nxyz + t] = s / (float)S;
}

// ---------------------------------------------------------------------------
// Host orchestration
// ---------------------------------------------------------------------------
extern "C" void kernel_launch(void* const* d_in, const int* in_sizes, int n_in,
                              void* d_out, int out_size, void* d_ws, size_t ws_size,
                              hipStream_t stream) {
    (void)in_sizes; (void)n_in; (void)out_size; (void)ws_size;
    const int B = 4, N0 = 16384;
    const int   NP[4]  = {2048, 1024, 512, 256};
    const int   NS[4]  = {64, 32, 16, 16};
    const float RAD[4] = {0.2f, 0.4f, 0.8f, 1.2f};
    const int   CIN0[4]    = {6, 131, 259, 259};
    const int   COUT[4][3] = {{64, 64, 128}, {128, 128, 256},
                              {128, 128, 256}, {256, 256, 512}};

    const float* pc = (const float*)d_in[0];
    const float *Wp[4][3], *Gp[4][3], *Bp[4][3];
    {
        int p = 1;
        for (int m = 0; m < 4; ++m)
            for (int l = 0; l < 3; ++l) {
                Wp[m][l] = (const float*)d_in[p++];
                Gp[m][l] = (const float*)d_in[p++];
                Bp[m][l] = (const float*)d_in[p++];
            }
    }

    // Workspace carving (256B aligned).
    char* cur_ws = (char*)d_ws;
    auto alloc = [&](size_t bytes) -> void* {
        void* r = (void*)cur_ws;
        cur_ws += (bytes + 255) & ~(size_t)255;
        return r;
    };
    float* mind  = (float*)alloc((size_t)B * N0 * sizeof(float));
    int*   inds  = (int*)  alloc((size_t)B * 2048 * sizeof(int));
    float* nxBuf[2];
    nxBuf[0] = (float*)alloc((size_t)B * 2048 * 3 * sizeof(float));
    nxBuf[1] = (float*)alloc((size_t)B * 2048 * 3 * sizeof(float));
    int*   bidx  = (int*)  alloc((size_t)B * 2048 * 64 * sizeof(int));
    float* fBuf[2];
    fBuf[0] = (float*)alloc((size_t)B * 256 * 2048 * sizeof(float));
    fBuf[1] = (float*)alloc((size_t)B * 256 * 2048 * sizeof(float));
    float* bufA = (float*)alloc((size_t)64  * 524288 * sizeof(float)); // 134 MB
    float* bufB = (float*)alloc((size_t)128 * 524288 * sizeof(float)); // 268 MB
    float* csum  = (float*)alloc(512 * sizeof(float));
    float* csq   = (float*)alloc(512 * sizeof(float));
    float* scale = (float*)alloc(512 * sizeof(float));
    float* shift = (float*)alloc(512 * sizeof(float));

    const float* xyzSrc = pc;   int xyzStride = 6;   int N = N0;
    const float* featsSrc = nullptr; // module 0 uses pointcloud channels 3..5
    float* lastNx = nullptr; float* lastFeats = nullptr;

    for (int m = 0; m < 4; ++m) {
        const int S = NP[m], K = NS[m];
        const long long ncol = (long long)B * S * K;
        float* nx = nxBuf[m & 1];

        // 1) FPS + centroid gather + ball query
        fps_kernel<<<B, 256, 0, stream>>>(xyzSrc, N, xyzStride, S, inds, mind);
        gather_xyz_kernel<<<(B * S + 255) / 256, 256, 0, stream>>>(
            xyzSrc, N, xyzStride, inds, S, B, nx);
        ball_query_kernel<<<(B * S + 255) / 256, 256, 0, stream>>>(
            xyzSrc, N, xyzStride, nx, S, B, RAD[m] * RAD[m], K, bidx);

        // 2) Materialize X (padded to multiple of 4 channels)
        const int cin0 = CIN0[m];
        const int Cf = cin0 - 3;
        const int cpad0 = (cin0 + 3) & ~3;
        const float* fbase; long long fBatch, fChan, fPt;
        if (m == 0) { fbase = pc + 3; fBatch = (long long)N0 * 6; fChan = 1; fPt = 6; }
        else        { fbase = featsSrc; fBatch = (long long)Cf * N; fChan = N; fPt = 1; }
        build_x_kernel<<<(unsigned)((ncol + 255) / 256), 256, 0, stream>>>(
            xyzSrc, N, xyzStride, nx, bidx, S, K, B, 1.0f / RAD[m],
            fbase, fBatch, fChan, fPt, Cf, bufA, cpad0);

        // 3) MLP chain: WMMA GEMM (register M-blocked) -> batch-stats BN -> ReLU.
        float* src = bufA; int cin = cin0, cpad = cpad0;
        for (int l = 0; l < 3; ++l) {
            const int O = COUT[m][l];
            float* dst = (l & 1) ? bufA : bufB;
            const int mtiles = O / 16;
            const int mt = mtiles < 8 ? mtiles : 8;
            fill_kernel<<<(O + 255) / 256, 256, 0, stream>>>(csum, O, 0.0f);
            fill_kernel<<<(O + 255) / 256, 256, 0, stream>>>(csq,  O, 0.0f);
            dim3 grid((unsigned)(ncol / 64), (unsigned)(mtiles / mt));
            if (mt == 8)
                gemm_bn_stats_kernel<8><<<grid, 128, 0, stream>>>(
                    Wp[m][l], cin, cpad, src, ncol, dst, csum, csq);
            else
                gemm_bn_stats_kernel<4><<<grid, 128, 0, stream>>>(
                    Wp[m][l], cin, cpad, src, ncol, dst, csum, csq);
            bn_finalize_kernel<<<(O + 63) / 64, 64, 0, stream>>>(
                csum, csq, Gp[m][l], Bp[m][l], O, 1.0f / (float)ncol, scale, shift);
            dim3 bgrid((unsigned)(ncol / 256), (unsigned)O);
            apply_bn_relu_kernel<<<bgrid, 256, 0, stream>>>(dst, ncol, scale, shift);
            src = dst; cin = O; cpad = O;
        }

        // 4) Max-pool over neighbors -> feats (B, C, S)
        const int Co = COUT[m][2];
        float* fout = fBuf[m & 1];
        maxpool_kernel<<<(B * Co * S + 255) / 256, 256, 0, stream>>>(
            src, B, Co, S, K, fout);

        xyzSrc = nx; xyzStride = 3; N = S; featsSrc = fout;
        lastNx = nx; lastFeats = fout;
    }

    // Final: centroids of stage 4 + S-mean of stage-4 features.
    const int total = B * 256 * 3 + B * 512;
    final_out_kernel<<<(total + 255) / 256, 256, 0, stream>>>(
        lastNx, lastFeats, B, 256, 512, (float*)d_out);
}